// YOLOLoss_55370718380434
// MI455X (gfx1250) — compile-verified
//
#include <hip/hip_runtime.h>
#include <hip/hip_bf16.h>
#include <math.h>

// ---------------------------------------------------------------------------
// YOLO loss on gfx1250 (MI455X).
//  - Label-gated streaming: read conf_labels first; load 81-ch confidence row
//    only for positive anchors (~5%), channel 0 for pos|neg, locations for pos.
//  - BCE algebra: per-class loss == softplus(x_c) - [c==label]*x_c, objectness
//    from a single log_sigmoid (logsig(-x) = logsig(x) - x).
//  - Wave reduction via V_WMMA_F32_16X16X4_F32 against an all-ones B matrix:
//    5 chained f32 WMMAs broadcast the exact 32-lane sum to every lane.
//  - Per-block partials -> d_ws (plain stores, no atomics), 1-block finalize.
// ---------------------------------------------------------------------------

typedef __attribute__((ext_vector_type(2))) float v2f;
typedef __attribute__((ext_vector_type(4))) float f4v;
typedef __attribute__((ext_vector_type(8))) float v8f;

#define BETA_F (1.0f / 9.0f)

// Sum p across all 32 lanes of the wave; result broadcast to every lane.
// EXEC must be all ones (call only from reconverged code).
__device__ __forceinline__ float wave_sum32(float p) {
  v2f ones; ones[0] = 1.0f; ones[1] = 1.0f;      // B = 4x16 all-ones
  v2f a;    a[0] = p;      a[1] = 0.0f;          // A[m][0]=p[m], A[m][2]=p[m+16]
  v8f z = {0.0f, 0.0f, 0.0f, 0.0f, 0.0f, 0.0f, 0.0f, 0.0f};
  // d vgpr r: lanes0-15 = s[r], lanes16-31 = s[r+8], s[i] = p[i] + p[i+16]
  v8f d = __builtin_amdgcn_wmma_f32_16x16x4_f32(false, a, false, ones,
                                                (short)0, z, false, false);
  v8f acc = {0.0f, 0.0f, 0.0f, 0.0f, 0.0f, 0.0f, 0.0f, 0.0f};
  v2f t;
  t[0] = d[0]; t[1] = d[1];   // contributes s0+s1+s8+s9 to every element
  acc = __builtin_amdgcn_wmma_f32_16x16x4_f32(false, t, false, ones,
                                              (short)0, acc, false, false);
  t[0] = d[2]; t[1] = d[3];
  acc = __builtin_amdgcn_wmma_f32_16x16x4_f32(false, t, false, ones,
                                              (short)0, acc, false, false);
  t[0] = d[4]; t[1] = d[5];
  acc = __builtin_amdgcn_wmma_f32_16x16x4_f32(false, t, false, ones,
                                              (short)0, acc, false, false);
  t[0] = d[6]; t[1] = d[7];
  acc = __builtin_amdgcn_wmma_f32_16x16x4_f32(false, t, false, ones,
                                              (short)0, acc, false, false);
  return acc[0];                 // all 256 elements equal the full 32-lane sum
}

// log(1 + exp(-|x|)) with fast HW transcendentals
__device__ __forceinline__ float l1pe_nabs(float x) {
  return __logf(1.0f + __expf(-fabsf(x)));
}

__global__ __launch_bounds__(256) void yolo_loss_main(
    const float* __restrict__ conf,
    const float* __restrict__ ploc,
    const float* __restrict__ gloc,
    const int*   __restrict__ labels,
    float*       __restrict__ ws,
    int N, int C1, int nBlocks)
{
  const int idx = blockIdx.x * 256 + threadIdx.x;

  float cnt = 0.0f;   // positive count
  float loc = 0.0f;   // smooth-L1 sum
  float cls = 0.0f;   // per-class BCE sum
  float bp  = 0.0f;   // objectness BCE, positive anchors
  float bn  = 0.0f;   // objectness BCE, negative anchors

  if (idx < N) {
    const int l = labels[idx];
    if (l >= 0) {                                   // -1 = ignore
      const float* row = conf + (size_t)idx * (size_t)C1;
      const float x0  = __builtin_nontemporal_load(row);
      const float ls0 = fminf(x0, 0.0f) - l1pe_nabs(x0);   // log_sigmoid(x0)
      if (l > 0) {                                   // positive anchor
        cnt = 1.0f;
        bp  = -ls0;
        float s  = 0.0f;
        float xl = 0.0f;
        #pragma unroll 4
        for (int c = 1; c < C1; ++c) {
          const float x = __builtin_nontemporal_load(row + c);
          s += fmaxf(x, 0.0f) + l1pe_nabs(x);        // softplus(x)
          xl = (c == l) ? x : xl;
        }
        cls = s - xl;                                // sum softplus - x_label

        const f4v p4 = __builtin_nontemporal_load((const f4v*)ploc + idx);
        const f4v g4 = __builtin_nontemporal_load((const f4v*)gloc + idx);
        #pragma unroll
        for (int k = 0; k < 4; ++k) {
          const float n = fabsf(p4[k] - g4[k]);
          loc += (n < BETA_F) ? (0.5f / BETA_F) * n * n
                              : n - 0.5f * BETA_F;
        }
      } else {                                       // negative anchor
        bn = x0 - ls0;                               // -log_sigmoid(-x0)
      }
    }
  }

  // ---- reconverged: full-EXEC WMMA wave reductions ----
  const float r0 = wave_sum32(cnt);
  const float r1 = wave_sum32(loc);
  const float r2 = wave_sum32(cls);
  const float r3 = wave_sum32(bp);
  const float r4 = wave_sum32(bn);

  __shared__ float smem[8][5];
  const int wave = threadIdx.x >> 5;
  const int lane = threadIdx.x & 31;
  if (lane == 0) {
    smem[wave][0] = r0; smem[wave][1] = r1; smem[wave][2] = r2;
    smem[wave][3] = r3; smem[wave][4] = r4;
  }
  __syncthreads();
  if (threadIdx.x < 5) {
    float t = 0.0f;
    #pragma unroll
    for (int w = 0; w < 8; ++w) t += smem[w][threadIdx.x];
    ws[threadIdx.x * nBlocks + blockIdx.x] = t;      // SoA per-block partials
  }
}

__global__ __launch_bounds__(256) void yolo_loss_finalize(
    const float* __restrict__ ws, float* __restrict__ out, int nBlocks)
{
  float a0 = 0.0f, a1 = 0.0f, a2 = 0.0f, a3 = 0.0f, a4 = 0.0f;
  for (int b = threadIdx.x; b < nBlocks; b += 256) {
    a0 += ws[0 * nBlocks + b];
    a1 += ws[1 * nBlocks + b];
    a2 += ws[2 * nBlocks + b];
    a3 += ws[3 * nBlocks + b];
    a4 += ws[4 * nBlocks + b];
  }
  const float r0 = wave_sum32(a0);
  const float r1 = wave_sum32(a1);
  const float r2 = wave_sum32(a2);
  const float r3 = wave_sum32(a3);
  const float r4 = wave_sum32(a4);

  __shared__ float smem[8][5];
  const int wave = threadIdx.x >> 5;
  const int lane = threadIdx.x & 31;
  if (lane == 0) {
    smem[wave][0] = r0; smem[wave][1] = r1; smem[wave][2] = r2;
    smem[wave][3] = r3; smem[wave][4] = r4;
  }
  __syncthreads();
  if (threadIdx.x == 0) {
    float cnt = 0.0f, loc = 0.0f, cls = 0.0f, bp = 0.0f, bn = 0.0f;
    #pragma unroll
    for (int w = 0; w < 8; ++w) {
      cnt += smem[w][0]; loc += smem[w][1]; cls += smem[w][2];
      bp  += smem[w][3]; bn  += smem[w][4];
    }
    const float np = fmaxf(1.0f, cnt);
    out[0] = loc / (np * 4.0f);                              // localisation
    const float classification = cls / np;
    const float binary = (bp * 1.0f + bn * 0.5f) / np;       // POS/NEG weights
    out[1] = 0.5f * (classification + binary);
  }
}

extern "C" void kernel_launch(void* const* d_in, const int* in_sizes, int n_in,
                              void* d_out, int out_size, void* d_ws, size_t ws_size,
                              hipStream_t stream) {
  const float* conf   = (const float*)d_in[0];
  const float* ploc   = (const float*)d_in[1];
  const float* gloc   = (const float*)d_in[2];
  const int*   labels = (const int*)d_in[3];

  const int N  = in_sizes[3];              // B*A anchors
  const int C1 = in_sizes[0] / N;          // C+1 = 81 channels

  const int threads = 256;
  const int nBlocks = (N + threads - 1) / threads;   // 2048 for the ref shapes

  float* ws = (float*)d_ws;                // 5 * nBlocks floats (~40 KB)

  yolo_loss_main<<<nBlocks, threads, 0, stream>>>(conf, ploc, gloc, labels,
                                                  ws, N, C1, nBlocks);
  yolo_loss_finalize<<<1, 256, 0, stream>>>(ws, (float*)d_out, nBlocks);
}